// GL2Loss_75874892251747
// MI455X (gfx1250) — compile-verified
//
#include <hip/hip_runtime.h>
#include <cstdint>
#include <cstddef>

// ---------------------------------------------------------------------------
// GL2 loss: mean( (lap(u) + c*psi_r)^2 + (c*psi_i)^2 ),  c = 1 - |psi|^2
// B=16, C=2, H=W=1024, fp32. Memory-bound: 256 MiB reads -> ~11 us floor
// at 23.3 TB/s. Strategy: async global->LDS staging of u row-bands (CDNA5
// ASYNCcnt path), streamed b128 loads of psi, wave32 shuffle reduction,
// deterministic two-stage scalar reduce (no float atomics).
// ---------------------------------------------------------------------------

#define BDIM 256
constexpr int kB = 16, kH = 1024, kW = 1024;
constexpr int ROWS = 8, HALO = 2, SLOTS = ROWS + 2 * HALO;  // 12 rows staged
constexpr int STRIPS = kH / ROWS;                           // 128

// Builtin signature (from clang diagnostic): arg0 is v4i* in AS1 (global
// source), arg1 expected v4i* in AS3 (LDS dest), then imm offset + cpol.
typedef __attribute__((ext_vector_type(4))) int v4i;
typedef __attribute__((address_space(1))) v4i gas_v4i;  // global int4
typedef __attribute__((address_space(3))) v4i las_v4i;  // LDS int4

// Async copy 16 bytes/lane from global to LDS (gfx1250, ASYNCcnt-tracked).
// Numeric casts are valid: generic LDS pointers carry the wave-relative LDS
// offset in their low 32 bits; generic global pointers equal their 64-bit
// flat address.
__device__ __forceinline__ void async_copy_b128(const float* gsrc, const float* ldst) {
#if __has_builtin(__builtin_amdgcn_global_load_async_to_lds_b128)
  __builtin_amdgcn_global_load_async_to_lds_b128(
      (gas_v4i*)(uintptr_t)gsrc, (las_v4i*)(uint32_t)(uintptr_t)ldst, 0, 0);
#else
  asm volatile("global_load_async_to_lds_b128 %0, %1, off"
               :: "v"((uint32_t)(uintptr_t)ldst), "v"((uint64_t)(uintptr_t)gsrc)
               : "memory");
#endif
}

__device__ __forceinline__ void wait_async_zero() {
#if __has_builtin(__builtin_amdgcn_s_wait_asynccnt)
  __builtin_amdgcn_s_wait_asynccnt(0);
#else
  asm volatile("s_wait_asynccnt 0" ::: "memory");
#endif
}

// Second derivative matching jnp.gradient(jnp.gradient(f)) along one axis.
// Interior (2 <= i <= n-3): (f[i+2] - 2 f[i] + f[i-2]) / 4, plus exact
// one-sided expansions at i in {0, 1, n-2, n-1}.
__device__ __forceinline__ float d2grad(float fm2, float fm1, float f0,
                                        float fp1, float fp2, int i, int n) {
  if (i >= 2 && i <= n - 3) return 0.25f * (fp2 - 2.0f * f0 + fm2);
  if (i == 0)     return 0.5f * fp2 - fp1 + 0.5f * f0;
  if (i == 1)     return 0.25f * fp2 - 0.75f * f0 + 0.5f * fm1;
  if (i == n - 2) return 0.5f * fp1 - 0.75f * f0 + 0.25f * fm2;
  /* i == n-1 */  return 0.5f * f0 - fm1 + 0.5f * fm2;
}

__global__ __launch_bounds__(BDIM) void gl2_main(const float* __restrict__ u,
                                                 const float* __restrict__ psi,
                                                 float* __restrict__ ws) {
  // 2 channels x 12 rows x 1024 cols of fp32 = 96 KiB (of 320 KiB/WGP).
  __shared__ __align__(16) float sU[2][SLOTS][kW];
  __shared__ float sRed[BDIM / 32];

  const int t   = threadIdx.x;
  const int blk = blockIdx.x;
  const int b   = blk / STRIPS;
  const int r0  = (blk % STRIPS) * ROWS;
  const size_t HW = (size_t)kH * kW;

  // ---- Stage u band (both channels, rows r0-2 .. r0+ROWS+1, clamped) ----
  // Each of 256 lanes moves one 16B chunk per row: 24 async b128 ops/lane.
  for (int ch = 0; ch < 2; ++ch) {
    const float* base = u + ((size_t)b * 2 + ch) * HW;
    for (int s = 0; s < SLOTS; ++s) {
      int gr = r0 - HALO + s;
      gr = gr < 0 ? 0 : (gr >= kH ? kH - 1 : gr);
      async_copy_b128(base + (size_t)gr * kW + t * 4, &sU[ch][s][t * 4]);
    }
  }
  wait_async_zero();   // this wave's async copies landed in LDS
  __syncthreads();     // all waves' copies visible

  const float* psir = psi + (size_t)b * 2 * HW;
  const float* psii = psir + HW;

  float acc = 0.0f;
  const int j0 = t * 4;  // 4 consecutive columns per thread
  for (int rr = 0; rr < ROWS; ++rr) {
    const int i = r0 + rr;
    // psi has no reuse: stream it with wide b128 loads.
    const float4 pr4 = *reinterpret_cast<const float4*>(psir + (size_t)i * kW + j0);
    const float4 pi4 = *reinterpret_cast<const float4*>(psii + (size_t)i * kW + j0);
    const float prs[4] = {pr4.x, pr4.y, pr4.z, pr4.w};
    const float pis[4] = {pi4.x, pi4.y, pi4.z, pi4.w};
#pragma unroll
    for (int e = 0; e < 4; ++e) {
      const int j   = j0 + e;
      const int jm2 = j - 2 < 0 ? 0 : j - 2;
      const int jm1 = j - 1 < 0 ? 0 : j - 1;
      const int jp1 = j + 1 >= kW ? kW - 1 : j + 1;
      const int jp2 = j + 2 >= kW ? kW - 1 : j + 2;
      float lap = 0.0f;
#pragma unroll
      for (int ch = 0; ch < 2; ++ch) {
        const float f0 = sU[ch][rr + 2][j];
        // vertical: LDS slots rr..rr+4 hold global rows i-2..i+2 (clamped
        // staging keeps the boundary formulas' operands valid)
        lap += d2grad(sU[ch][rr][j], sU[ch][rr + 1][j], f0,
                      sU[ch][rr + 3][j], sU[ch][rr + 4][j], i, kH);
        // horizontal
        lap += d2grad(sU[ch][rr + 2][jm2], sU[ch][rr + 2][jm1], f0,
                      sU[ch][rr + 2][jp1], sU[ch][rr + 2][jp2], j, kW);
      }
      const float pr = prs[e], pi = pis[e];
      const float c  = 1.0f - (pr * pr + pi * pi);
      const float re = lap + c * pr;
      const float im = c * pi;
      acc = fmaf(re, re, fmaf(im, im, acc));
    }
  }

  // ---- wave32 reduction, then per-block partial ----
#pragma unroll
  for (int o = 16; o > 0; o >>= 1) acc += __shfl_xor(acc, o, 32);
  if ((t & 31) == 0) sRed[t >> 5] = acc;
  __syncthreads();
  if (t == 0) {
    float ssum = 0.0f;
#pragma unroll
    for (int w = 0; w < BDIM / 32; ++w) ssum += sRed[w];
    ws[blk] = ssum;  // deterministic per-block partial
  }
}

// Deterministic final reduction of per-block partials -> scalar mean.
__global__ __launch_bounds__(BDIM) void gl2_reduce(const float* __restrict__ ws,
                                                   float* __restrict__ out,
                                                   int n, float scale) {
  __shared__ float s[BDIM];
  const int t = threadIdx.x;
  float a = 0.0f;
  for (int idx = t; idx < n; idx += BDIM) a += ws[idx];
  s[t] = a;
  __syncthreads();
  for (int o = BDIM / 2; o > 0; o >>= 1) {
    if (t < o) s[t] += s[t + o];
    __syncthreads();
  }
  if (t == 0) out[0] = s[0] * scale;
}

extern "C" void kernel_launch(void* const* d_in, const int* in_sizes, int n_in,
                              void* d_out, int out_size, void* d_ws, size_t ws_size,
                              hipStream_t stream) {
  (void)in_sizes; (void)n_in; (void)out_size; (void)ws_size;
  const float* u   = (const float*)d_in[0];
  const float* psi = (const float*)d_in[1];
  float* out = (float*)d_out;
  float* ws  = (float*)d_ws;   // needs kB*STRIPS*4 = 8 KiB of scratch

  const int nblk = kB * STRIPS;  // 2048 blocks, 8 wave32s each
  gl2_main<<<dim3(nblk), dim3(BDIM), 0, stream>>>(u, psi, ws);
  const float scale = 1.0f / ((float)kB * (float)kH * (float)kW);
  gl2_reduce<<<dim3(1), dim3(BDIM), 0, stream>>>(ws, out, nblk, scale);
}